// CoDMO_41068477284684
// MI455X (gfx1250) — compile-verified
//
#include <hip/hip_runtime.h>
#include <hip/hip_bf16.h>
#include <math.h>

// ---------------- constants from the reference ----------------
#define EDIM 128
#define ADIM 128
#define G3E  384
#define NLD  12000
#define NDT  13000
#define NLP  8000
#define NPT  8800
#define BB   128
#define TT   40
#define CD   30
#define CP   20

typedef float v2f __attribute__((ext_vector_type(2)));
typedef float v8f __attribute__((ext_vector_type(8)));

// =====================================================================
// WMMA f32 GEMM:  C[M,N] = gather(Atab, idx)[M,128] @ op(B)[128,N] + bias
//  - GATHER: A row i = Atab[idx[i*idx_stride]], else Atab row i directly
//  - BT=0: B row-major [128,N];  BT=1: B is W[N,128], computes @ W^T
// One wave (32 lanes) computes one 16x16 tile with V_WMMA_F32_16X16X4_F32.
// Branch-free inner loop: pointer+stride addressing, row masking by 0-mul.
// =====================================================================
template <bool BT, bool GATHER>
__global__ void gemm_wmma_f32(const float* __restrict__ Atab,
                              const int* __restrict__ idx, int idx_stride,
                              int M, const float* __restrict__ B, int N,
                              const float* __restrict__ bias,
                              float* __restrict__ C) {
  const int K = EDIM;
  int lane = threadIdx.x & 31;      // 0..31
  int half = lane >> 4;             // 0 / 1
  int l15  = lane & 15;
  int i0 = blockIdx.x * 16;
  int j0 = blockIdx.y * 16;

  int rowA   = i0 + l15;
  float am   = (rowA < M) ? 1.0f : 0.0f;   // zero out-of-range rows (no branch)
  int rclamp = (rowA < M) ? rowA : 0;
  long arow  = GATHER ? (long)idx[(long)rclamp * idx_stride] : (long)rclamp;
  const float* Ar = Atab + arow * (long)K;
  int jn = j0 + l15;                 // N is always a multiple of 16

  const float* Bp = BT ? (B + (long)jn * K) : (B + jn);
  const long bstr = BT ? 1 : (long)N;

  v8f acc = {0.f, 0.f, 0.f, 0.f, 0.f, 0.f, 0.f, 0.f};
#pragma unroll 8
  for (int k0 = 0; k0 < K; k0 += 4) {
    int ka = k0 + half * 2;
    v2f a, b;
    a.x = Ar[ka] * am;
    a.y = Ar[ka + 1] * am;
    b.x = Bp[(long)ka * bstr];
    b.y = Bp[(long)(ka + 1) * bstr];
    acc = __builtin_amdgcn_wmma_f32_16x16x4_f32(false, a, false, b,
                                                (short)0, acc, false, false);
  }
  float bv = bias ? bias[jn] : 0.0f;
#pragma unroll
  for (int r = 0; r < 8; ++r) {
    int row = i0 + r + half * 8;
    if (row < M) C[(long)row * N + jn] = acc[r] + bv;
  }
}

static void launch_gemm(const float* Atab, const int* idx, int idx_stride, int M,
                        const float* B, int N, int Btrans, const float* bias,
                        float* C, hipStream_t s) {
  dim3 grid((M + 15) / 16, N / 16);
  if (Btrans) {
    if (idx) gemm_wmma_f32<true,  true ><<<grid, 32, 0, s>>>(Atab, idx, idx_stride, M, B, N, bias, C);
    else     gemm_wmma_f32<true,  false><<<grid, 32, 0, s>>>(Atab, nullptr, 1, M, B, N, bias, C);
  } else {
    if (idx) gemm_wmma_f32<false, true ><<<grid, 32, 0, s>>>(Atab, idx, idx_stride, M, B, N, bias, C);
    else     gemm_wmma_f32<false, false><<<grid, 32, 0, s>>>(Atab, nullptr, 1, M, B, N, bias, C);
  }
}

// =====================================================================
// Leaf attention epilogue: one block (128 threads) per leaf g.
// pre[m] = dot(leakyrelu(Q[g] + P[nb[g,m]] + ba), va) + mask[g,m]
// attn   = softmax_m(pre) * softmax_m(w)
// out[g] = sum_m attn[m] * W[nb[g,m]]
// =====================================================================
__global__ void leaf_combine_kernel(const float* __restrict__ Q, const float* __restrict__ P,
                                    const int* __restrict__ nb, const float* __restrict__ mask,
                                    const float* __restrict__ wgt,
                                    const float* __restrict__ ba, const float* __restrict__ va,
                                    const float* __restrict__ W,
                                    float* __restrict__ outLeaf) {
  int g = blockIdx.x;
  int t = threadIdx.x;              // 0..127 == E index
  __shared__ float red[128];
  __shared__ float pre[16];
  __shared__ float attn[16];
  const int* nbg = nb + (long)g * 16;
  float qv  = Q[(long)g * ADIM + t] + ba[t];
  float vav = va[t];
  for (int m = 0; m < 16; ++m) {
    float x = qv + P[(long)nbg[m] * ADIM + t];
    x = (x > 0.f) ? x : 0.01f * x;          // jax.nn.leaky_relu default slope
    red[t] = x * vav;
    __syncthreads();
    for (int s = 64; s > 0; s >>= 1) {
      if (t < s) red[t] += red[t + s];
      __syncthreads();
    }
    if (t == 0) pre[m] = red[0] + mask[(long)g * 16 + m];
    __syncthreads();
  }
  if (t == 0) {
    float mx1 = -1e30f, mx2 = -1e30f;
    for (int m = 0; m < 16; ++m) {
      mx1 = fmaxf(mx1, pre[m]);
      mx2 = fmaxf(mx2, wgt[(long)g * 16 + m]);
    }
    float s1 = 0.f, s2 = 0.f, ea[16], eb[16];
    for (int m = 0; m < 16; ++m) {
      ea[m] = expf(pre[m] - mx1);                     s1 += ea[m];
      eb[m] = expf(wgt[(long)g * 16 + m] - mx2);      s2 += eb[m];
    }
    for (int m = 0; m < 16; ++m) attn[m] = (ea[m] / s1) * (eb[m] / s2);
  }
  __syncthreads();
  float acc = 0.f;
  for (int m = 0; m < 16; ++m) acc += attn[m] * W[(long)nbg[m] * EDIM + t];
  outLeaf[(long)g * EDIM + t] = acc;
}

// =====================================================================
// p2c epilogue: one block per parent g in a group starting at row par0.
// out[g] = 0.5 * ( sum_m softmax(pre)[m]*W[ch[g,m]] + W[par0+g] )
// =====================================================================
__global__ void p2c_combine_kernel(const float* __restrict__ Qp, const float* __restrict__ Pc,
                                   const int* __restrict__ ch, const float* __restrict__ mask,
                                   const float* __restrict__ ba, const float* __restrict__ va,
                                   const float* __restrict__ W, int par0,
                                   float* __restrict__ tmpOut) {
  int g = blockIdx.x;
  int t = threadIdx.x;
  __shared__ float red[128];
  __shared__ float pre[16];
  __shared__ float attn[16];
  const int* chg = ch + (long)g * 16;
  float qv  = Qp[(long)g * ADIM + t] + ba[t];
  float vav = va[t];
  for (int m = 0; m < 16; ++m) {
    float x = qv + Pc[((long)g * 16 + m) * ADIM + t];
    x = (x > 0.f) ? x : 0.01f * x;
    red[t] = x * vav;
    __syncthreads();
    for (int s = 64; s > 0; s >>= 1) {
      if (t < s) red[t] += red[t + s];
      __syncthreads();
    }
    if (t == 0) pre[m] = red[0] + mask[(long)g * 16 + m];
    __syncthreads();
  }
  if (t == 0) {
    float mx = -1e30f;
    for (int m = 0; m < 16; ++m) mx = fmaxf(mx, pre[m]);
    float s = 0.f, ea[16];
    for (int m = 0; m < 16; ++m) { ea[m] = expf(pre[m] - mx); s += ea[m]; }
    for (int m = 0; m < 16; ++m) attn[m] = ea[m] / s;
  }
  __syncthreads();
  float acc = 0.f;
  for (int m = 0; m < 16; ++m) acc += attn[m] * W[(long)chg[m] * EDIM + t];
  float pe = W[((long)par0 + g) * EDIM + t];
  tmpOut[(long)g * EDIM + t] = 0.5f * (acc + pe);
}

// GRU gate fusion (torch gate order r,z,n):  h' = (1-z)*n + z*h
__global__ void gru_gate_kernel(const float* __restrict__ gi, const float* __restrict__ gh,
                                float* __restrict__ h, int G) {
  long i = (long)blockIdx.x * blockDim.x + threadIdx.x;
  if (i >= (long)G * EDIM) return;
  long g = i >> 7;
  int  e = (int)(i & 127);
  const float* gig = gi + g * G3E;
  const float* ghg = gh + g * G3E;
  float r = 1.f / (1.f + expf(-(gig[e]       + ghg[e])));
  float z = 1.f / (1.f + expf(-(gig[128 + e] + ghg[128 + e])));
  float n = tanhf(gig[256 + e] + r * ghg[256 + e]);
  h[i] = (1.f - z) * n + z * h[i];
}

// c2p write-back: W[leaf] = 0.5*(h + W[leaf])
__global__ void c2p_write_kernel(const float* __restrict__ h, float* __restrict__ W, int G) {
  long i = (long)blockIdx.x * blockDim.x + threadIdx.x;
  if (i >= (long)G * EDIM) return;
  W[i] = 0.5f * (h[i] + W[i]);
}

// visit: out[bt, base+e] = tanh( sum_c cm[bt,c] * (2*e2[x]+e3[x])[e] )
__global__ void visit_kernel(const float* __restrict__ e2leaf, const float* __restrict__ W,
                             const int* __restrict__ x, const float* __restrict__ cm,
                             int C, float* __restrict__ out, int base) {
  int bt = blockIdx.x;        // 0 .. B*T-1 (row-major over [B,T])
  int e  = threadIdx.x;       // 0 .. 127
  float acc = 0.f;
  for (int c = 0; c < C; ++c) {
    long xi = x[(long)bt * C + c];
    float L = 2.f * e2leaf[xi * EDIM + e] + W[xi * EDIM + e];
    acc += cm[(long)bt * C + c] * L;
  }
  out[(long)bt * 256 + base + e] = tanhf(acc);
}

// =====================================================================
// Host orchestration
// =====================================================================
namespace {

struct OntArgs {
  const float* W0; int Ntot; int nleaf;
  const float *lf1W, *lf1b, *lf1v, *lf2W, *lf2b, *lf2v;
  const float *attW, *attb, *attv, *Wih, *Whh, *bih, *bhh;
  const int* nb[2]; const float* msk[2]; const float* wgt[2];
  int par0[3]; int gsz[3];
  const int* pch[3]; const float* pmsk[3];
  const int* cpar;
  // scratch
  float *W, *e2, *Q, *P, *newLeaf, *Qp, *Pc, *ptmp, *gi, *gh, *h;
};

static void run_leaf_attn(const OntArgs& o, const float* argTab,
                          const float* lw, const float* lb, const float* lv,
                          hipStream_t s) {
  // Q = argTab[0:nleaf] @ Wa_top     (same for both iterations)
  launch_gemm(argTab, nullptr, 1, o.nleaf, lw, ADIM, 0, nullptr, o.Q, s);
  for (int it = 0; it < 2; ++it) {
    // P = W[0:nleaf] @ Wa_bot
    launch_gemm(o.W, nullptr, 1, o.nleaf, lw + EDIM * ADIM, ADIM, 0, nullptr, o.P, s);
    leaf_combine_kernel<<<o.nleaf, 128, 0, s>>>(
        o.Q, o.P, o.nb[it], o.msk[it], o.wgt[it], lb, lv, o.W, o.newLeaf);
    hipMemcpyAsync(o.W, o.newLeaf, (size_t)o.nleaf * EDIM * sizeof(float),
                   hipMemcpyDeviceToDevice, s);
  }
}

static void run_ont(const OntArgs& o, const int* x, const float* cm, int C, int base,
                    float* out, hipStream_t s) {
  // W_tmp = copy(W0)
  hipMemcpyAsync(o.W, o.W0, (size_t)o.Ntot * EDIM * sizeof(float),
                 hipMemcpyDeviceToDevice, s);
  // ---- e1 = leaf_attn(W0, leaf1) ---- (node_e reads the ORIGINAL table)
  run_leaf_attn(o, o.W0, o.lf1W, o.lf1b, o.lf1v, s);

  // ---- tree attention: p2c groups (sequential) ----
  for (int j = 0; j < 3; ++j) {
    int p0 = o.par0[j], gs = o.gsz[j];
    launch_gemm(o.W + (long)p0 * EDIM, nullptr, 1, gs, o.attW, ADIM, 0, nullptr, o.Qp, s);
    launch_gemm(o.W, o.pch[j], 1, gs * 16, o.attW + EDIM * ADIM, ADIM, 0, nullptr, o.Pc, s);
    p2c_combine_kernel<<<gs, 128, 0, s>>>(
        o.Qp, o.Pc, o.pch[j], o.pmsk[j], o.attb, o.attv, o.W, p0, o.ptmp);
    hipMemcpyAsync(o.W + (long)p0 * EDIM, o.ptmp, (size_t)gs * EDIM * sizeof(float),
                   hipMemcpyDeviceToDevice, s);
  }
  // ---- c2p: 4-step GRU over ancestor path ----
  hipMemsetAsync(o.h, 0, (size_t)o.nleaf * EDIM * sizeof(float), s);
  int eltBlocks = (o.nleaf * EDIM + 255) / 256;
  for (int t = 0; t < 4; ++t) {
    // gi = gather(W, par[:,t]) @ Wih^T + bih
    launch_gemm(o.W, o.cpar + t, 4, o.nleaf, o.Wih, G3E, 1, o.bih, o.gi, s);
    // gh = h @ Whh^T + bhh
    launch_gemm(o.h, nullptr, 1, o.nleaf, o.Whh, G3E, 1, o.bhh, o.gh, s);
    gru_gate_kernel<<<eltBlocks, 256, 0, s>>>(o.gi, o.gh, o.h, o.nleaf);
  }
  c2p_write_kernel<<<eltBlocks, 256, 0, s>>>(o.h, o.W, o.nleaf);

  // ---- save e2 leaf rows (leaves are rows [0, nleaf)) ----
  hipMemcpyAsync(o.e2, o.W, (size_t)o.nleaf * EDIM * sizeof(float),
                 hipMemcpyDeviceToDevice, s);
  // ---- e3 = leaf_attn(e2, leaf2) ----
  run_leaf_attn(o, o.e2, o.lf2W, o.lf2b, o.lf2v, s);
  // ---- visit: tanh(sum_c cm * (2*e2 + e3)) ----
  visit_kernel<<<BB * TT, 128, 0, s>>>(o.e2, o.W, x, cm, C, out, base);
}

} // namespace

extern "C" void kernel_launch(void* const* d_in, const int* in_sizes, int n_in,
                              void* d_out, int out_size, void* d_ws, size_t ws_size,
                              hipStream_t stream) {
  (void)in_sizes; (void)n_in; (void)out_size; (void)ws_size;

  // ---- input map (setup_inputs dict order, tuples flattened) ----
  const int*   x_diag   = (const int*)  d_in[0];
  const float* cm_diag  = (const float*)d_in[2];
  const int*   x_proc   = (const int*)  d_in[3];
  const float* cm_proc  = (const float*)d_in[5];
  const float* Wemb_d   = (const float*)d_in[6];
  const float* Wemb_p   = (const float*)d_in[7];
  const float* lf1W = (const float*)d_in[8],  *lf1b = (const float*)d_in[9],  *lf1v = (const float*)d_in[10];
  const float* lf2W = (const float*)d_in[11], *lf2b = (const float*)d_in[12], *lf2v = (const float*)d_in[13];
  const float* attW = (const float*)d_in[14], *attb = (const float*)d_in[15], *attv = (const float*)d_in[16];
  const float* Wih  = (const float*)d_in[17], *Whh  = (const float*)d_in[18];
  const float* bih  = (const float*)d_in[19], *bhh  = (const float*)d_in[20];
  // diag structures: 21,22 nodes (unused); 23,24 neighbors; 25,26 masks; 27,28 weights
  // 29-31 p2c parents (unused, offsets hardcoded); 32-34 children; 35-37 masks; 38 c2p par; 39 c2p child
  // proc: 40,41 nodes; 42,43 neighbors; 44,45 masks; 46,47 weights; 48-50 par; 51-53 child; 54-56 mask; 57 c2p par; 58 c2p child

  // ---- workspace carve-up (floats) ----
  float* w = (float*)d_ws;
  auto take = [&](size_t n) { float* p = w; w += n; return p; };
  float* W_d   = take((size_t)NDT * EDIM);
  float* W_p   = take((size_t)NPT * EDIM);
  float* e2_d  = take((size_t)NLD * EDIM);
  float* e2_p  = take((size_t)NLP * EDIM);
  float* Qb    = take((size_t)NLD * ADIM);
  float* Pb    = take((size_t)NLD * ADIM);
  float* newL  = take((size_t)NLD * EDIM);
  float* Qp    = take((size_t)512 * ADIM);
  float* Pc    = take((size_t)8192 * ADIM);
  float* ptmp  = take((size_t)512 * EDIM);
  float* gi    = take((size_t)NLD * G3E);
  float* gh    = take((size_t)NLD * G3E);
  float* hb    = take((size_t)NLD * EDIM);

  OntArgs od{};
  od.W0 = Wemb_d; od.Ntot = NDT; od.nleaf = NLD;
  od.lf1W = lf1W; od.lf1b = lf1b; od.lf1v = lf1v;
  od.lf2W = lf2W; od.lf2b = lf2b; od.lf2v = lf2v;
  od.attW = attW; od.attb = attb; od.attv = attv;
  od.Wih = Wih; od.Whh = Whh; od.bih = bih; od.bhh = bhh;
  od.nb[0] = (const int*)d_in[23]; od.nb[1] = (const int*)d_in[24];
  od.msk[0] = (const float*)d_in[25]; od.msk[1] = (const float*)d_in[26];
  od.wgt[0] = (const float*)d_in[27]; od.wgt[1] = (const float*)d_in[28];
  od.par0[0] = 12000; od.par0[1] = 12100; od.par0[2] = 12500;
  od.gsz[0] = 100; od.gsz[1] = 400; od.gsz[2] = 500;
  od.pch[0] = (const int*)d_in[32]; od.pch[1] = (const int*)d_in[33]; od.pch[2] = (const int*)d_in[34];
  od.pmsk[0] = (const float*)d_in[35]; od.pmsk[1] = (const float*)d_in[36]; od.pmsk[2] = (const float*)d_in[37];
  od.cpar = (const int*)d_in[38];
  od.W = W_d; od.e2 = e2_d; od.Q = Qb; od.P = Pb; od.newLeaf = newL;
  od.Qp = Qp; od.Pc = Pc; od.ptmp = ptmp; od.gi = gi; od.gh = gh; od.h = hb;

  OntArgs op = od;
  op.W0 = Wemb_p; op.Ntot = NPT; op.nleaf = NLP;
  op.nb[0] = (const int*)d_in[42]; op.nb[1] = (const int*)d_in[43];
  op.msk[0] = (const float*)d_in[44]; op.msk[1] = (const float*)d_in[45];
  op.wgt[0] = (const float*)d_in[46]; op.wgt[1] = (const float*)d_in[47];
  op.par0[0] = 8000; op.par0[1] = 8080; op.par0[2] = 8400;
  op.gsz[0] = 80; op.gsz[1] = 320; op.gsz[2] = 400;
  op.pch[0] = (const int*)d_in[51]; op.pch[1] = (const int*)d_in[52]; op.pch[2] = (const int*)d_in[53];
  op.pmsk[0] = (const float*)d_in[54]; op.pmsk[1] = (const float*)d_in[55]; op.pmsk[2] = (const float*)d_in[56];
  op.cpar = (const int*)d_in[57];
  op.W = W_p; op.e2 = e2_p;

  float* out = (float*)d_out;
  run_ont(od, x_diag, cm_diag, CD, 0,   out, stream);
  run_ont(op, x_proc, cm_proc, CP, 128, out, stream);
}